// GraphSAGE_12429635355189
// MI455X (gfx1250) — compile-verified
//
#include <hip/hip_runtime.h>
#include <hip/hip_bf16.h>

// GraphSAGE forward for MI455X (gfx1250, wave32).
//  - edge aggregation: float4 scatter-atomics (L2-resident gathers: all feature
//    matrices fit in the 192MB L2)
//  - SAGE linear layers: V_WMMA_F32_16X16X4_F32. One wave owns one 16-row node
//    tile and ALL FOUT/16 output tiles (multi-accumulator), so each A fragment
//    (agg*1/deg and x) is loaded once per K-step and reused across every output
//    tile -> ~2x fewer vector loads per WMMA in the wide layers.
//  - BN stats via per-channel atomics in the GEMM epilogue; BN+ELU (+residual /
//    dual-output for layer 4) as a fused elementwise pass.

typedef float v2f __attribute__((ext_vector_type(2)));
typedef float v8f __attribute__((ext_vector_type(8)));

#define BN_EPS 1e-5f

// ---------------------------------------------------------------- degree
__global__ void degree_kernel(const long long* __restrict__ dst,
                              float* __restrict__ deg, long long E) {
  long long e = (long long)blockIdx.x * blockDim.x + threadIdx.x;
  if (e < E) atomicAdd(&deg[(int)dst[e]], 1.0f);
}

__global__ void invdeg_kernel(float* __restrict__ deg, int n) {
  int i = blockIdx.x * blockDim.x + threadIdx.x;
  if (i < n) deg[i] = 1.0f / fmaxf(deg[i], 1.0f);   // jnp.maximum(deg, 1)
}

// ------------------------------------------------- edge scatter (segment_sum)
// idx = e*C4 + c4 with c4 fastest -> one wave covers consecutive float4 of one
// edge row: coalesced gather, adjacent-address atomics (combine in L2).
template <int FIN>
__global__ void scatter_kernel(const float* __restrict__ x,
                               const long long* __restrict__ src,
                               const long long* __restrict__ dst,
                               float* __restrict__ agg, long long E) {
  const int C4 = FIN / 4;
  long long idx = (long long)blockIdx.x * blockDim.x + threadIdx.x;
  if (idx >= E * (long long)C4) return;
  long long e = idx / C4;
  int c4 = (int)(idx - e * (long long)C4);
  int s = (int)src[e];
  int d = (int)dst[e];
  float4 v = ((const float4*)x)[(long long)s * C4 + c4];
  float* a = agg + (long long)d * FIN + c4 * 4;
  atomicAdd(a + 0, v.x);
  atomicAdd(a + 1, v.y);
  atomicAdd(a + 2, v.z);
  atomicAdd(a + 3, v.w);
}

// ---------------------------------------------------------- fused SAGE GEMM
// hpre[n,o] = sum_k (agg[n,k]*invdeg[n])*Wl[o,k] + x[n,k]*Wr[o,k]  + bl[o]
//
// f32 16x4 A layout (ISA 7.12.2): lane = m + 16*khalf; the 2 A VGPRs hold the
// K pair {2*khalf, 2*khalf+1}. B (4x16) mirrors it: lane = n + 16*khalf. Both
// fragments are 8-byte float2 loads from row-major storage.
//
// One wave: node tile tn (16 rows) x all TILES_O output tiles. A fragments are
// loaded once per K-step and reused for all 2*TILES_O WMMAs of that step.
template <int FIN, int FOUT>
__global__ void sage_gemm_kernel(const float* __restrict__ agg,
                                 const float* __restrict__ invdeg,
                                 const float* __restrict__ xin,
                                 const float* __restrict__ Wl,  // [FOUT,FIN]
                                 const float* __restrict__ bl,  // [FOUT]
                                 const float* __restrict__ Wr,  // [FOUT,FIN]
                                 float* __restrict__ hpre,      // [N,FOUT]
                                 float* __restrict__ stat_sum,  // [FOUT]
                                 float* __restrict__ stat_sq,   // [FOUT]
                                 int nrows) {
  constexpr int TILES_O = FOUT / 16;
  int wave = blockIdx.x * (blockDim.x >> 5) + (threadIdx.x >> 5);
  int lane = threadIdx.x & 31;
  int tiles_n = nrows >> 4;                       // N divisible by 16
  if (wave >= tiles_n) return;                    // wave-uniform: EXEC stays full
  int tn = wave;

  int m     = lane & 15;
  int khalf = lane >> 4;                          // 0: K={0,1}, 1: K={2,3}
  int row   = tn * 16 + m;                        // A fragment row

  const float* aggRow = agg + (long long)row * FIN;
  const float* xRow   = xin + (long long)row * FIN;
  const float* wlB    = Wl  + (long long)m * FIN; // B base (otile 0)
  const float* wrB    = Wr  + (long long)m * FIN;
  float idg = invdeg[row];

  v8f c[TILES_O];
#pragma unroll
  for (int t = 0; t < TILES_O; ++t)
    c[t] = (v8f){0.f, 0.f, 0.f, 0.f, 0.f, 0.f, 0.f, 0.f};

#pragma unroll 2
  for (int k0 = 0; k0 < FIN; k0 += 4) {
    int k = k0 + khalf * 2;                       // even -> 8B aligned
    float2 av = *(const float2*)(aggRow + k);
    float2 xv = *(const float2*)(xRow + k);
    v2f aagg; aagg.x = av.x * idg; aagg.y = av.y * idg;  // fold mean-division
    v2f ax;   ax.x   = xv.x;       ax.y   = xv.y;
#pragma unroll
    for (int t = 0; t < TILES_O; ++t) {
      float2 b1 = *(const float2*)(wlB + (long long)t * 16 * FIN + k);
      v2f b; b.x = b1.x; b.y = b1.y;
      c[t] = __builtin_amdgcn_wmma_f32_16x16x4_f32(false, aagg, false, b,
                                                   (short)0, c[t], false, false);
      float2 b2 = *(const float2*)(wrB + (long long)t * 16 * FIN + k);
      v2f bb; bb.x = b2.x; bb.y = b2.y;
      c[t] = __builtin_amdgcn_wmma_f32_16x16x4_f32(false, ax, false, bb,
                                                   (short)0, c[t], false, false);
    }
  }

  // Epilogue: bias, store, per-channel BN statistics.
#pragma unroll
  for (int t = 0; t < TILES_O; ++t) {
    int ocol = t * 16 + m;                        // D: lane holds column ocol
    float bias = bl[ocol];
    float s = 0.f, s2 = 0.f;
#pragma unroll
    for (int v = 0; v < 8; ++v) {                 // rows v + 8*khalf
      float val = c[t][v] + bias;
      int rowm = tn * 16 + v + 8 * khalf;
      hpre[(long long)rowm * FOUT + ocol] = val;
      s  += val;
      s2 += val * val;
    }
    atomicAdd(&stat_sum[ocol], s);
    atomicAdd(&stat_sq[ocol], s2);
  }
}

// ------------------------------------------------------ BN + ELU (+extras)
// bn = (h-mu)*rsqrt(var+eps)*gamma+beta ; act = elu(bn) (+residual)
// bn_out != null : also store bn pre-activation (layer 4's out_bn4)
template <int FOUT>
__global__ void bn_elu_kernel(const float* __restrict__ hpre,
                              const float* __restrict__ ssum,
                              const float* __restrict__ ssq,
                              const float* __restrict__ gamma,
                              const float* __restrict__ beta,
                              const float* __restrict__ residual,  // nullable
                              float* __restrict__ bn_out,          // nullable
                              float* __restrict__ act_out,
                              int nrows) {
  long long idx = (long long)blockIdx.x * blockDim.x + threadIdx.x;
  if (idx >= (long long)nrows * FOUT) return;
  int cfeat = (int)(idx % FOUT);
  float invN = 1.0f / (float)nrows;
  float mu   = ssum[cfeat] * invN;
  float var  = ssq[cfeat] * invN - mu * mu;       // biased batch variance
  float h    = hpre[idx];
  float bn   = (h - mu) * rsqrtf(var + BN_EPS) * gamma[cfeat] + beta[cfeat];
  if (bn_out) bn_out[idx] = bn;
  float a = bn > 0.f ? bn : expm1f(bn);           // ELU
  if (residual) a += residual[idx];               // h4_in = h3 + h1
  act_out[idx] = a;
}

// ------------------------------------------------------------- classifier
__global__ void classifier_kernel(const float* __restrict__ h4,  // [N,32]
                                  const float* __restrict__ Wc,  // [2,32]
                                  const float* __restrict__ bc,  // [2]
                                  float* __restrict__ logits, int nrows) {
  int n = blockIdx.x * blockDim.x + threadIdx.x;
  if (n >= nrows) return;
  const float4* r = (const float4*)(h4 + (long long)n * 32);
  float acc0 = bc[0], acc1 = bc[1];
#pragma unroll
  for (int i = 0; i < 8; ++i) {
    float4 v  = r[i];
    float4 w0 = ((const float4*)Wc)[i];
    float4 w1 = ((const float4*)(Wc + 32))[i];
    acc0 += v.x * w0.x + v.y * w0.y + v.z * w0.z + v.w * w0.w;
    acc1 += v.x * w1.x + v.y * w1.y + v.z * w1.z + v.w * w1.w;
  }
  logits[(long long)n * 2 + 0] = acc0;
  logits[(long long)n * 2 + 1] = acc1;
}

// ---------------------------------------------------------------- launch
extern "C" void kernel_launch(void* const* d_in, const int* in_sizes, int n_in,
                              void* d_out, int out_size, void* d_ws, size_t ws_size,
                              hipStream_t stream) {
  (void)n_in; (void)out_size; (void)ws_size;

  const float*     x     = (const float*)d_in[0];
  const long long* edges = (const long long*)d_in[1];
  const int       N = in_sizes[0] / 128;
  const long long E = (long long)in_sizes[1] / 2;
  const long long* src = edges;
  const long long* dst = edges + E;

  const float *W1l=(const float*)d_in[2],  *b1l=(const float*)d_in[3],
              *W1r=(const float*)d_in[4],  *g1 =(const float*)d_in[5],
              *be1=(const float*)d_in[6];
  const float *W2l=(const float*)d_in[7],  *b2l=(const float*)d_in[8],
              *W2r=(const float*)d_in[9],  *g2 =(const float*)d_in[10],
              *be2=(const float*)d_in[11];
  const float *W3l=(const float*)d_in[12], *b3l=(const float*)d_in[13],
              *W3r=(const float*)d_in[14], *g3 =(const float*)d_in[15],
              *be3=(const float*)d_in[16];
  const float *W4l=(const float*)d_in[17], *b4l=(const float*)d_in[18],
              *W4r=(const float*)d_in[19], *g4 =(const float*)d_in[20],
              *be4=(const float*)d_in[21];
  const float *Wc =(const float*)d_in[22], *bc =(const float*)d_in[23];

  // workspace layout (floats): ~167 MB total
  float* ws     = (float*)d_ws;
  float* invdeg = ws;                           // N
  float* agg    = invdeg + (size_t)N;           // N*128 (reused per layer)
  float* buf1   = agg  + (size_t)N * 128;       // N*64  : hpre1 -> h1 (in place)
  float* buf2   = buf1 + (size_t)N * 64;        // N*128 : hpre2 -> h2 (in place)
  float* buf3   = buf2 + (size_t)N * 128;       // N*64  : hpre3 -> h4_in (in place)
  float* h4     = buf3 + (size_t)N * 64;        // N*32
  float* ssum   = h4   + (size_t)N * 32;        // 128
  float* ssq    = ssum + 128;                   // 128

  // output layout: logits[N,2] | out_conv4[N,32] | out_bn4[N,32]
  float* logits = (float*)d_out;
  float* conv4  = logits + (size_t)N * 2;
  float* bn4    = conv4 + (size_t)N * 32;

  auto cdiv = [](long long a, long long b) { return (unsigned)((a + b - 1) / b); };
  const unsigned gemm_blocks = cdiv(N / 16, 4);   // 4 waves/block, 1 node-tile/wave

  // degrees (shared by all layers): deg -> 1/max(deg,1)
  hipMemsetAsync(invdeg, 0, (size_t)N * 4, stream);
  degree_kernel<<<cdiv(E, 256), 256, 0, stream>>>(dst, invdeg, E);
  invdeg_kernel<<<cdiv(N, 256), 256, 0, stream>>>(invdeg, N);

  // ---- layer 1: 128 -> 64, input x, act -> buf1
  hipMemsetAsync(agg, 0, (size_t)N * 128 * 4, stream);
  scatter_kernel<128><<<cdiv(E * 32, 256), 256, 0, stream>>>(x, src, dst, agg, E);
  hipMemsetAsync(ssum, 0, 256 * 4, stream);
  sage_gemm_kernel<128, 64><<<gemm_blocks, 128, 0, stream>>>(
      agg, invdeg, x, W1l, b1l, W1r, buf1, ssum, ssq, N);
  bn_elu_kernel<64><<<cdiv((long long)N * 64, 256), 256, 0, stream>>>(
      buf1, ssum, ssq, g1, be1, nullptr, nullptr, buf1, N);

  // ---- layer 2: 64 -> 128, input h1(buf1), act -> buf2
  hipMemsetAsync(agg, 0, (size_t)N * 64 * 4, stream);
  scatter_kernel<64><<<cdiv(E * 16, 256), 256, 0, stream>>>(buf1, src, dst, agg, E);
  hipMemsetAsync(ssum, 0, 256 * 4, stream);
  sage_gemm_kernel<64, 128><<<gemm_blocks, 128, 0, stream>>>(
      agg, invdeg, buf1, W2l, b2l, W2r, buf2, ssum, ssq, N);
  bn_elu_kernel<128><<<cdiv((long long)N * 128, 256), 256, 0, stream>>>(
      buf2, ssum, ssq, g2, be2, nullptr, nullptr, buf2, N);

  // ---- layer 3: 128 -> 64, input h2(buf2), act+residual(h1) -> buf3 = h4_in
  hipMemsetAsync(agg, 0, (size_t)N * 128 * 4, stream);
  scatter_kernel<128><<<cdiv(E * 32, 256), 256, 0, stream>>>(buf2, src, dst, agg, E);
  hipMemsetAsync(ssum, 0, 256 * 4, stream);
  sage_gemm_kernel<128, 64><<<gemm_blocks, 128, 0, stream>>>(
      agg, invdeg, buf2, W3l, b3l, W3r, buf3, ssum, ssq, N);
  bn_elu_kernel<64><<<cdiv((long long)N * 64, 256), 256, 0, stream>>>(
      buf3, ssum, ssq, g3, be3, /*residual=*/buf1, nullptr, buf3, N);

  // ---- layer 4: 64 -> 32, input h4_in(buf3); hpre -> conv4 (d_out),
  //      bn -> bn4 (d_out), act -> h4
  hipMemsetAsync(agg, 0, (size_t)N * 64 * 4, stream);
  scatter_kernel<64><<<cdiv(E * 16, 256), 256, 0, stream>>>(buf3, src, dst, agg, E);
  hipMemsetAsync(ssum, 0, 256 * 4, stream);
  sage_gemm_kernel<64, 32><<<gemm_blocks, 128, 0, stream>>>(
      agg, invdeg, buf3, W4l, b4l, W4r, conv4, ssum, ssq, N);
  bn_elu_kernel<32><<<cdiv((long long)N * 32, 256), 256, 0, stream>>>(
      conv4, ssum, ssq, g4, be4, nullptr, /*bn_out=*/bn4, h4, N);

  // ---- classifier: logits = h4 @ Wc.T + bc
  classifier_kernel<<<cdiv(N, 256), 256, 0, stream>>>(h4, Wc, bc, logits, N);
}